// WCT_9895604650749
// MI455X (gfx1250) — compile-verified
//
#include <hip/hip_runtime.h>
#include <hip/hip_bf16.h>
#include <math.h>

// ---------------- problem constants (from reference) ----------------
#define NG        64      // G*B independent items
#define NMEM      64      // channels per group (n_mem)
#define HW_N      4096    // H*W
#define NS_ITERS  10
#define EPS_F     1e-5f
#define LSTR      68      // padded LDS row stride in floats -> conflict-free 16-lane column reads
#define KSPLIT    4       // K-dimension split of the covariance GEMM (parallelism)

typedef float v2f __attribute__((ext_vector_type(2)));
typedef float v8f __attribute__((ext_vector_type(8)));

// ---------------- async global->LDS (CDNA5 ASYNCcnt path), guarded ----------------
#if defined(__has_builtin)
#if __has_builtin(__builtin_amdgcn_global_load_async_to_lds_b128) && \
    __has_builtin(__builtin_amdgcn_s_wait_asynccnt)
#define HAVE_ASYNC_LDS 1
#endif
#endif
#ifndef HAVE_ASYNC_LDS
#define HAVE_ASYNC_LDS 0
#endif

#if HAVE_ASYNC_LDS
typedef int v4i_gcc __attribute__((vector_size(4 * sizeof(int))));
typedef __attribute__((address_space(1))) v4i_gcc* g4_t;   // global int4*
typedef __attribute__((address_space(3))) v4i_gcc* l4_t;   // LDS int4*
#endif

__device__ __forceinline__ void async_cp16(const float* g, float* l) {
#if HAVE_ASYNC_LDS
  __builtin_amdgcn_global_load_async_to_lds_b128((g4_t)g, (l4_t)l, 0, 0);
#else
  *(float4*)l = *(const float4*)g;
#endif
}

template <int N>
__device__ __forceinline__ void wait_async() {
#if HAVE_ASYNC_LDS
  __builtin_amdgcn_s_wait_asynccnt(N);
#endif
}

// ---------------- fp32 WMMA 16x16x4 wrapper ----------------
__device__ __forceinline__ v8f wmma4(v2f a, v2f b, v8f c) {
#if defined(__has_builtin) && __has_builtin(__builtin_amdgcn_wmma_f32_16x16x4_f32)
  return __builtin_amdgcn_wmma_f32_16x16x4_f32(false, a, false, b, (short)0, c, false, false);
#else
  c[0] += a.x * b.x;  // compile-safety fallback only
  return c;
#endif
}

// Symmetric-output tile schedule for a 4x4 grid of 16x16 tiles (10 unique tiles, 8 waves):
//   waves 0..3 -> diagonal tiles (w,w); waves 4..7 -> (0,1),(0,2),(0,3),(1,2);
//   waves 2,3 additionally own (1,3),(2,3)  (keeps per-SIMD tile count at 3 vs 4 dense).
__device__ __forceinline__ void sym_tiles(int wv, int& i0, int& j0, bool& has1, int& i1, int& j1) {
  if (wv < 4) { i0 = wv; j0 = wv; }
  else if (wv == 4) { i0 = 0; j0 = 1; }
  else if (wv == 5) { i0 = 0; j0 = 2; }
  else if (wv == 6) { i0 = 0; j0 = 3; }
  else { i0 = 1; j0 = 2; }
  has1 = (wv == 2) || (wv == 3);
  i1 = (wv == 2) ? 1 : 2;
  j1 = 3;
}

// Symmetric 64x64x64 LDS matmul: Ds = diagAdd*I + scale*(As @ Bs), result assumed symmetric
// (As, Bs commuting symmetric matrices). Off-diagonal tiles are mirror-stored.
__device__ __forceinline__ void mm_sym(const float* As, const float* Bs, float* Ds,
                                       int wv, int lane, float scale, float diagAdd) {
  int i0, j0, i1, j1; bool has1;
  sym_tiles(wv, i0, j0, has1, i1, j1);
  const int rsel = lane & 15;
  const int khalf = 2 * (lane >> 4);
  const v8f vzero = {0.f, 0.f, 0.f, 0.f, 0.f, 0.f, 0.f, 0.f};
  v8f acc0 = vzero, acc1 = vzero;
  if (has1) {
#pragma unroll
    for (int ks = 0; ks < 16; ++ks) {
      const int kb = ks * 4 + khalf;
      v2f a0 = *(const v2f*)&As[(i0 * 16 + rsel) * LSTR + kb];
      v2f b0; b0.x = Bs[kb * LSTR + j0 * 16 + rsel]; b0.y = Bs[(kb + 1) * LSTR + j0 * 16 + rsel];
      acc0 = wmma4(a0, b0, acc0);
      v2f a1 = *(const v2f*)&As[(i1 * 16 + rsel) * LSTR + kb];
      v2f b1; b1.x = Bs[kb * LSTR + j1 * 16 + rsel]; b1.y = Bs[(kb + 1) * LSTR + j1 * 16 + rsel];
      acc1 = wmma4(a1, b1, acc1);
    }
  } else {
#pragma unroll
    for (int ks = 0; ks < 16; ++ks) {
      const int kb = ks * 4 + khalf;
      v2f a0 = *(const v2f*)&As[(i0 * 16 + rsel) * LSTR + kb];
      v2f b0; b0.x = Bs[kb * LSTR + j0 * 16 + rsel]; b0.y = Bs[(kb + 1) * LSTR + j0 * 16 + rsel];
      acc0 = wmma4(a0, b0, acc0);
    }
  }
#pragma unroll
  for (int r = 0; r < 8; ++r) {
    const int row = i0 * 16 + r + 8 * (lane >> 4);
    const int col = j0 * 16 + rsel;
    float v = scale * acc0[r];
    if (row == col) v += diagAdd;
    Ds[row * LSTR + col] = v;
    if (i0 != j0) Ds[col * LSTR + row] = v;
  }
  if (has1) {
#pragma unroll
    for (int r = 0; r < 8; ++r) {
      const int row = i1 * 16 + r + 8 * (lane >> 4);
      const int col = j1 * 16 + rsel;
      const float v = scale * acc1[r];
      Ds[row * LSTR + col] = v;
      Ds[col * LSTR + row] = v;  // i1 != j1 always
    }
  }
}

// ---------------- kernel 1: partial Gram matrices + partial row sums ----------------
// grid = 2*NG*KSPLIT = 512 blocks. Block handles 1024 K-columns; emits raw partial S
// (full 64x64 via symmetric mirror) and 64 partial row sums. Double-buffered async staging.
__global__ __launch_bounds__(256) void wct_cov(const float* __restrict__ cA,
                                               const float* __restrict__ sB,
                                               float* __restrict__ Spart,
                                               float* __restrict__ rsumPart) {
  __shared__ float tileA[64 * LSTR];
  __shared__ float tileB[64 * LSTR];
  __shared__ float rred[64 * 4];

  const int bid = blockIdx.x;
  const int split = bid & (KSPLIT - 1);
  const int pk = bid >> 2;                 // kind*64 + p
  const int p = pk & 63;
  const int kind = pk >> 6;
  const float* X = (kind == 0 ? cA : sB) + (size_t)p * NMEM * HW_N + split * (HW_N / KSPLIT);

  const int t = threadIdx.x;
  const int lane = t & 31, wv = t >> 5;
  const int lr = t >> 2, lseg = t & 3;     // row-sum read mapping
  const int rsel = lane & 15, khalf = 2 * (lane >> 4);

  int i0, j0, i1, j1; bool has1;
  sym_tiles(wv, i0, j0, has1, i1, j1);

  // stage mapping: thread t copies 4 float4 slots s = t + q*256 -> (row = s/16, seg16B = s%16)
  const v8f vzero = {0.f, 0.f, 0.f, 0.f, 0.f, 0.f, 0.f, 0.f};
  v8f acc0 = vzero, acc1 = vzero;
  float rowacc = 0.f;

  const int NCH = (HW_N / KSPLIT) / 64;    // 16 chunks of 64 columns

  // prologue: issue chunk 0 into tileA
#pragma unroll
  for (int q = 0; q < 4; ++q) {
    const int s = t + q * 256, row = s >> 4, seg = s & 15;
    async_cp16(X + (size_t)row * HW_N + seg * 4, &tileA[row * LSTR + seg * 4]);
  }

  for (int c = 0; c < NCH; ++c) {
    float* cur = (c & 1) ? tileB : tileA;
    float* nxt = (c & 1) ? tileA : tileB;
    if (c + 1 < NCH) {
      const float* Xc = X + (c + 1) * 64;
#pragma unroll
      for (int q = 0; q < 4; ++q) {
        const int s = t + q * 256, row = s >> 4, seg = s & 15;
        async_cp16(Xc + (size_t)row * HW_N + seg * 4, &nxt[row * LSTR + seg * 4]);
      }
      wait_async<4>();   // in-order: chunk c's 4 loads complete, chunk c+1 may be in flight
    } else {
      wait_async<0>();
    }
    __syncthreads();     // all waves' chunk-c data visible in LDS

    // fold row sums from LDS (thread t: row lr, 16 cols at lseg*16)
    {
      const float* rp = cur + lr * LSTR + lseg * 16;
#pragma unroll
      for (int q = 0; q < 4; ++q) {
        float4 v = *(const float4*)(rp + q * 4);
        rowacc += v.x + v.y + v.z + v.w;
      }
    }

    // S += X_tile @ X_tile^T on the 10 unique tiles
    if (has1) {
#pragma unroll
      for (int ks = 0; ks < 16; ++ks) {
        const int kb = ks * 4 + khalf;
        v2f a0 = *(const v2f*)&cur[(i0 * 16 + rsel) * LSTR + kb];
        v2f b0 = *(const v2f*)&cur[(j0 * 16 + rsel) * LSTR + kb];
        acc0 = wmma4(a0, b0, acc0);
        v2f a1 = *(const v2f*)&cur[(i1 * 16 + rsel) * LSTR + kb];
        v2f b1 = *(const v2f*)&cur[(j1 * 16 + rsel) * LSTR + kb];
        acc1 = wmma4(a1, b1, acc1);
      }
    } else {
#pragma unroll
      for (int ks = 0; ks < 16; ++ks) {
        const int kb = ks * 4 + khalf;
        v2f a0 = *(const v2f*)&cur[(i0 * 16 + rsel) * LSTR + kb];
        v2f b0 = *(const v2f*)&cur[(j0 * 16 + rsel) * LSTR + kb];
        acc0 = wmma4(a0, b0, acc0);
      }
    }
    __syncthreads();     // all reads of `cur` done before it is overwritten
  }

  // partial row sums
  rred[lr * 4 + lseg] = rowacc;
  __syncthreads();
  if (t < 64)
    rsumPart[(size_t)(pk * KSPLIT + split) * 64 + t] =
        rred[t * 4] + rred[t * 4 + 1] + rred[t * 4 + 2] + rred[t * 4 + 3];

  // raw partial S (mirror off-diagonal tiles)
  float* S = Spart + (size_t)(pk * KSPLIT + split) * NMEM * NMEM;
#pragma unroll
  for (int r = 0; r < 8; ++r) {
    const int row = i0 * 16 + r + 8 * (lane >> 4);
    const int col = j0 * 16 + rsel;
    S[row * NMEM + col] = acc0[r];
    if (i0 != j0) S[col * NMEM + row] = acc0[r];
  }
  if (has1) {
#pragma unroll
    for (int r = 0; r < 8; ++r) {
      const int row = i1 * 16 + r + 8 * (lane >> 4);
      const int col = j1 * 16 + rsel;
      S[row * NMEM + col] = acc1[r];
      S[col * NMEM + row] = acc1[r];
    }
  }
}

// ---------------- kernel 2: reduce partials -> cov, then coupled Newton-Schulz ----------------
// grid 128: bx<64 -> content (emit cov^{-1/2}), else style (emit cov^{1/2}); also emits means.
__global__ __launch_bounds__(256) void wct_ns(const float* __restrict__ Spart,
                                              const float* __restrict__ rsumPart,
                                              float* __restrict__ meanC, float* __restrict__ meanB,
                                              float* __restrict__ isqC, float* __restrict__ sqB) {
  __shared__ float bufs[5][64 * LSTR];
  __shared__ float red[256];
  __shared__ float meansh[64];

  const int p = blockIdx.x & 63;
  const int wantInv = (blockIdx.x >> 6) == 0;
  const int pk = (wantInv ? 0 : 64) + p;
  const float* Sp = Spart + (size_t)pk * KSPLIT * NMEM * NMEM;
  const float* Rp = rsumPart + (size_t)pk * KSPLIT * 64;
  float* outM = (wantInv ? isqC : sqB) + (size_t)p * NMEM * NMEM;
  float* meanOut = (wantInv ? meanC : meanB) + (size_t)p * NMEM;

  const int t = threadIdx.x, lane = t & 31, wv = t >> 5;

  if (t < 64) {
    const float s = Rp[t] + Rp[64 + t] + Rp[128 + t] + Rp[192 + t];
    const float m = s * (1.f / (float)HW_N);
    meansh[t] = m;
    meanOut[t] = m;
  }
  __syncthreads();

  // cov = (sum_k Spart_k - HW*m*m^T)/(HW-1) + eps*I ; accumulate Frobenius norm
  const float inv = 1.f / (float)(HW_N - 1);
  float ss = 0.f;
  for (int e = t; e < NMEM * NMEM; e += 256) {
    const int r = e >> 6, c = e & 63;
    float s = Sp[e] + Sp[4096 + e] + Sp[8192 + e] + Sp[12288 + e];
    float cv = (s - (float)HW_N * meansh[r] * meansh[c]) * inv;
    if (r == c) cv += EPS_F;
    bufs[0][r * LSTR + c] = cv;
    ss += cv * cv;
  }
  red[t] = ss;
  __syncthreads();
  if (t < 64) red[t] = red[t] + red[t + 64] + red[t + 128] + red[t + 192];
  __syncthreads();
  if (t == 0) {
    float s = 0.f;
    for (int q = 0; q < 64; ++q) s += red[q];
    red[0] = sqrtf(s);
  }
  __syncthreads();
  const float normA = red[0];
  const float rn = 1.f / normA;

  // Y = A/||A||, Z = I
  for (int e = t; e < NMEM * NMEM; e += 256) {
    const int r = e >> 6, c = e & 63;
    bufs[0][r * LSTR + c] *= rn;
    bufs[1][r * LSTR + c] = (r == c) ? 1.f : 0.f;
  }
  __syncthreads();

  int iY = 0, iZ = 1, iT = 2, iU = 3, iV = 4;
  for (int it = 0; it < NS_ITERS; ++it) {
    mm_sym(bufs[iZ], bufs[iY], bufs[iT], wv, lane, -0.5f, 1.5f);   // T = 1.5I - 0.5 Z@Y
    __syncthreads();
    mm_sym(bufs[iY], bufs[iT], bufs[iU], wv, lane, 1.f, 0.f);      // newY = Y@T
    mm_sym(bufs[iT], bufs[iZ], bufs[iV], wv, lane, 1.f, 0.f);      // newZ = T@Z
    __syncthreads();
    int s;
    s = iY; iY = iU; iU = s;
    s = iZ; iZ = iV; iV = s;
  }

  const float sc = wantInv ? (1.f / sqrtf(normA)) : sqrtf(normA);
  const float* R = bufs[wantInv ? iZ : iY];
  for (int e = t; e < NMEM * NMEM; e += 256) {
    const int r = e >> 6, c = e & 63;
    outM[e] = R[r * LSTR + c] * sc;
  }
}

// ---------------- kernel 3: M = cov_b^{1/2} @ cov_c^{-1/2} (symmetric product) ----------------
__global__ __launch_bounds__(256) void wct_combine(const float* __restrict__ sqB,
                                                   const float* __restrict__ isqC,
                                                   float* __restrict__ Mm) {
  __shared__ float As[64 * LSTR];
  __shared__ float Bs[64 * LSTR];
  __shared__ float Ds[64 * LSTR];
  const int p = blockIdx.x;
  const int t = threadIdx.x, lane = t & 31, wv = t >> 5;
  const float* A = sqB + (size_t)p * NMEM * NMEM;
  const float* B = isqC + (size_t)p * NMEM * NMEM;
  for (int e = t; e < NMEM * NMEM; e += 256) {
    const int r = e >> 6, c = e & 63;
    As[r * LSTR + c] = A[e];
    Bs[r * LSTR + c] = B[e];
  }
  __syncthreads();
  mm_sym(As, Bs, Ds, wv, lane, 1.f, 0.f);
  __syncthreads();
  for (int e = t; e < NMEM * NMEM; e += 256) {
    const int r = e >> 6, c = e & 63;
    Mm[(size_t)p * NMEM * NMEM + e] = Ds[r * LSTR + c];
  }
}

// ---------------- kernel 4: coloring GEMM + fused epilogue ----------------
// out = alpha*(M @ cA + corr) + (1-alpha)*cA,  corr[r] = m_b[r] - (M @ m_c)[r]
// grid 512: p = bx/8, column chunk = bx%8 (512 cols); each wave handles 64 cols.
__global__ __launch_bounds__(256) void wct_color(const float* __restrict__ cA,
                                                 const float* __restrict__ Mm,
                                                 const float* __restrict__ meanC,
                                                 const float* __restrict__ meanB,
                                                 const float* __restrict__ alphaPtr,
                                                 float* __restrict__ out) {
  __shared__ float Ms[64 * LSTR];
  __shared__ float mcs[64];
  __shared__ float corr[64];
  __shared__ float red2[256];

  const int p = blockIdx.x >> 3;
  const int chunk = blockIdx.x & 7;
  const int t = threadIdx.x, lane = t & 31, wv = t >> 5;
  const int rsel = lane & 15, khalf = 2 * (lane >> 4);
  const float* Ap = cA + (size_t)p * NMEM * HW_N;

  const float alpha = alphaPtr[0];
  const float beta = 1.f - alpha;

  for (int e = t; e < NMEM * NMEM; e += 256)
    Ms[(e >> 6) * LSTR + (e & 63)] = Mm[(size_t)p * NMEM * NMEM + e];
  if (t < 64) mcs[t] = meanC[p * NMEM + t];
  __syncthreads();

  {  // corr = m_b - M @ m_c
    const int r = t & 63, q = t >> 6;
    float sp = 0.f;
#pragma unroll
    for (int k = 0; k < 16; ++k) sp += Ms[r * LSTR + q * 16 + k] * mcs[q * 16 + k];
    red2[t] = sp;
  }
  __syncthreads();
  if (t < 64) corr[t] = meanB[p * NMEM + t] - (red2[t] + red2[t + 64] + red2[t + 128] + red2[t + 192]);
  __syncthreads();

  const int n0 = chunk * 512 + wv * 64;
  const v8f vzero = {0.f, 0.f, 0.f, 0.f, 0.f, 0.f, 0.f, 0.f};

  for (int nt = 0; nt < 4; ++nt) {
    const int cn = n0 + nt * 16;
    const int col = cn + rsel;
    v8f acc[4];
#pragma unroll
    for (int i = 0; i < 4; ++i) acc[i] = vzero;

#pragma unroll
    for (int ks = 0; ks < 16; ++ks) {
      const int kb = ks * 4 + khalf;
      v2f b;
      b.x = Ap[(size_t)kb * HW_N + col];
      b.y = Ap[(size_t)(kb + 1) * HW_N + col];
#pragma unroll
      for (int i = 0; i < 4; ++i) {
        v2f a = *(const v2f*)&Ms[(i * 16 + rsel) * LSTR + kb];
        acc[i] = wmma4(a, b, acc[i]);
      }
    }

#pragma unroll
    for (int i = 0; i < 4; ++i) {
#pragma unroll
      for (int r = 0; r < 8; ++r) {
        const int row = i * 16 + r + 8 * (lane >> 4);
        const size_t idx = (size_t)p * NMEM * HW_N + (size_t)row * HW_N + col;
        const float orig = cA[idx];
        out[idx] = alpha * (acc[i][r] + corr[row]) + beta * orig;
      }
    }
  }
}

// ---------------- launcher ----------------
extern "C" void kernel_launch(void* const* d_in, const int* in_sizes, int n_in,
                              void* d_out, int out_size, void* d_ws, size_t ws_size,
                              hipStream_t stream) {
  const float* cA = (const float*)d_in[0];
  const float* sB = (const float*)d_in[1];
  const float* alpha = (const float*)d_in[2];
  float* out = (float*)d_out;

  // workspace carving (floats): Spart 2M + rsum 32K + 3 result arrays + means  ~= 11.7 MB
  float* ws = (float*)d_ws;
  const size_t MAT = (size_t)NG * NMEM * NMEM;          // 262144 floats
  float* Spart = ws;                                    // 2*NG*KSPLIT*64*64 = 2,097,152
  float* rsumP = Spart + (size_t)2 * NG * KSPLIT * NMEM * NMEM;
  float* isq_c = rsumP + (size_t)2 * NG * KSPLIT * NMEM;  // 32768
  float* sq_b  = isq_c + MAT;
  float* Mm    = sq_b + MAT;
  float* mean_c = Mm + MAT;
  float* mean_b = mean_c + (size_t)NG * NMEM;
  (void)in_sizes; (void)n_in; (void)out_size; (void)ws_size;

  wct_cov<<<2 * NG * KSPLIT, 256, 0, stream>>>(cA, sB, Spart, rsumP);
  wct_ns<<<128, 256, 0, stream>>>(Spart, rsumP, mean_c, mean_b, isq_c, sq_b);
  wct_combine<<<64, 256, 0, stream>>>(sq_b, isq_c, Mm);
  wct_color<<<512, 256, 0, stream>>>(cA, Mm, mean_c, mean_b, alpha, out);
}